// GAT_23124103922017
// MI455X (gfx1250) — compile-verified
//
#include <hip/hip_runtime.h>
#include <math.h>

// ============================================================================
// 2-layer single-head GAT (PyG GATConv semantics, with self-loops) for MI455X.
//
// Per layer:
//   1) h = x @ W        -> V_WMMA_F32_16X16X4_F32 (exact fp32), B staged in LDS
//   2) a_s[i]=h[i].att_src, a_d[i]=h[i].att_dst; init max=-inf, denom=0, agg=0
//   3) e = leaky_relu(a_s[src]+a_d[dst]); atomicMax(max[dst], e)   (E+N items)
//   4) p = exp(e - max[dst]); atomicAdd(denom[dst], p); store p
//   5) wave32-per-edge: agg[dst] += h[src] * (p/denom[dst])  (coalesced rows)
//   6) out = agg + bias (+ ReLU after layer 1)
// ============================================================================

#define GAT_NEG_SLOPE 0.2f

typedef float v2f __attribute__((ext_vector_type(2)));
typedef float v8f __attribute__((ext_vector_type(8)));

// ---------------------------------------------------------------------------
// GEMM via V_WMMA_F32_16X16X4_F32.  C(M x NP) = A(M x K) * B(K x Nv), with
// NP = NT*16 >= Nv (C is stored column-padded; pad cols receive zeros).
//
// B is staged once per block into LDS, zero-padded, pair-interleaved so the
// hot loop's B fragment is a single conflict-free ds_load_b64:
//   ldsB[ ((k>>1)*NP + col)*2 + (k&1) ] = B[k][col]
//
// Fragment layouts (per CDNA5 ISA 7.12.2):
//   A 16x4 f32 : lanes 0-15 row=lane, K{0,1}; lanes 16-31 row=lane-16, K{2,3}
//                -> contiguous float2 per lane (8B aligned, K % 4 == 0).
//   B 4x16 f32 : lanes 0-15 col=lane, rows K{0,1}; lanes 16-31 col=lane-16,
//                rows K{2,3} -> one v2f from the interleaved LDS image.
//   C/D 16x16  : VGPR r = row r (lanes 0-15) / row r+8 (lanes 16-31).
// ---------------------------------------------------------------------------
template <int NT, int KMAX>
__global__ void gat_gemm_wmma(const float* __restrict__ A,
                              const float* __restrict__ B,
                              float* __restrict__ C,
                              int M, int K, int Nv) {
  constexpr int NP = NT * 16;
  __shared__ float ldsB[KMAX * NP];

  // ---- cold staging: whole B -> LDS, zero-pad cols [Nv, NP) ----
  const int KNP = K * NP;
  for (int i = threadIdx.x; i < KNP; i += blockDim.x) {
    const int kk = i / NP;
    const int c  = i - kk * NP;
    const float v = (c < Nv) ? B[(size_t)kk * Nv + c] : 0.0f;
    ldsB[(((kk >> 1) * NP + c) << 1) + (kk & 1)] = v;
  }
  __syncthreads();

  const int wave = (int)((blockIdx.x * blockDim.x + threadIdx.x) >> 5);
  const int lane = threadIdx.x & 31;
  const int m0 = wave * 16;
  if (m0 >= M) return;

  const int lh  = lane & 15;
  const int khi = (lane >> 4) << 1;  // 0 for lanes 0-15, 2 for lanes 16-31

  int arow = m0 + lh;
  if (arow >= M) arow = M - 1;       // tail guard; stores are row-masked below
  const float* __restrict__ Ar = A + (size_t)arow * K;

  v8f acc[NT] = {};

  for (int k = 0; k < K; k += 4) {
    const v2f a = *reinterpret_cast<const v2f*>(Ar + k + khi);     // global b64
    const int kp = (k + khi) >> 1;                                  // pair index
#pragma unroll
    for (int t = 0; t < NT; ++t) {
      const v2f b = *reinterpret_cast<const v2f*>(
          &ldsB[(kp * NP + t * 16 + lh) << 1]);                     // ds b64
      acc[t] = __builtin_amdgcn_wmma_f32_16x16x4_f32(
          /*neg_a=*/false, a, /*neg_b=*/false, b,
          /*c_mod=*/(short)0, acc[t], /*reuse_a=*/false, /*reuse_b=*/false);
    }
  }

  const int rbase = m0 + ((lane >> 4) << 3);
  if (m0 + 16 <= M) {                 // fast path: fully in-range tile
#pragma unroll
    for (int t = 0; t < NT; ++t) {
      const int col = t * 16 + lh;
#pragma unroll
      for (int r = 0; r < 8; ++r)
        C[(size_t)(rbase + r) * NP + col] = acc[t][r];
    }
  } else {
#pragma unroll
    for (int t = 0; t < NT; ++t) {
      const int col = t * 16 + lh;
#pragma unroll
      for (int r = 0; r < 8; ++r) {
        const int row = rbase + r;
        if (row < M) C[(size_t)row * NP + col] = acc[t][r];
      }
    }
  }
}

// ---------------------------------------------------------------------------
// Per-node: attention logit terms + init of max / denom / aggregation buffer.
// h has row stride ldH (may be column-padded); agg has row stride F.
// ---------------------------------------------------------------------------
__global__ void gat_node_prep(const float* __restrict__ h, int ldH,
                              const float* __restrict__ att_src,
                              const float* __restrict__ att_dst,
                              float* __restrict__ a_s, float* __restrict__ a_d,
                              float* __restrict__ mx, float* __restrict__ dn,
                              float* __restrict__ agg, int N, int F) {
  const int i = blockIdx.x * blockDim.x + threadIdx.x;
  if (i >= N) return;
  const float* __restrict__ hr = h + (size_t)i * ldH;
  float s = 0.0f, d = 0.0f;
  for (int f = 0; f < F; ++f) {
    const float v = hr[f];
    s += v * att_src[f];
    d += v * att_dst[f];
  }
  a_s[i] = s;
  a_d[i] = d;
  mx[i] = -INFINITY;
  dn[i] = 0.0f;
  float* __restrict__ ar = agg + (size_t)i * F;
  for (int f = 0; f < F; ++f) ar[f] = 0.0f;
}

// Edge idx < E: real edge (src = ei[idx], dst = ei[E+idx]); otherwise self-loop.
__device__ __forceinline__ void gat_endpoints(const long long* __restrict__ ei,
                                              long long E, long long idx,
                                              int& s, int& d) {
  if (idx < E) {
    s = (int)ei[idx];
    d = (int)ei[E + idx];
  } else {
    s = d = (int)(idx - E);
  }
}

__device__ __forceinline__ float gat_leaky(float e) {
  return (e > 0.0f) ? e : GAT_NEG_SLOPE * e;
}

__global__ void gat_edge_max(const long long* __restrict__ ei,
                             const float* __restrict__ a_s,
                             const float* __restrict__ a_d,
                             float* __restrict__ mx, long long E, int N) {
  const long long idx = (long long)blockIdx.x * blockDim.x + threadIdx.x;
  if (idx >= E + N) return;
  int s, d;
  gat_endpoints(ei, E, idx, s, d);
  const float e = gat_leaky(a_s[s] + a_d[d]);
  atomicMax(&mx[d], e);  // global_atomic_max_num_f32
}

__global__ void gat_edge_exp(const long long* __restrict__ ei,
                             const float* __restrict__ a_s,
                             const float* __restrict__ a_d,
                             const float* __restrict__ mx,
                             float* __restrict__ dn, float* __restrict__ p,
                             long long E, int N) {
  const long long idx = (long long)blockIdx.x * blockDim.x + threadIdx.x;
  if (idx >= E + N) return;
  int s, d;
  gat_endpoints(ei, E, idx, s, d);
  const float e = gat_leaky(a_s[s] + a_d[d]);
  const float pv = expf(e - mx[d]);
  p[idx] = pv;
  atomicAdd(&dn[d], pv);
}

// One wave32 per edge: coalesced gather of h[src] row (stride ldH), scatter-add
// into agg[dst] row (stride F).
__global__ void gat_edge_aggregate(const long long* __restrict__ ei,
                                   const float* __restrict__ p,
                                   const float* __restrict__ dn,
                                   const float* __restrict__ h, int ldH,
                                   float* __restrict__ agg,
                                   long long E, int N, int F) {
  const long long wid = ((long long)blockIdx.x * blockDim.x + threadIdx.x) >> 5;
  const int lane = threadIdx.x & 31;
  if (wid >= E + N) return;
  int s, d;
  gat_endpoints(ei, E, wid, s, d);
  const float alpha = p[wid] / dn[d];
  const float* __restrict__ hr = h + (size_t)s * ldH;
  float* __restrict__ ar = agg + (size_t)d * F;
  for (int f = lane; f < F; f += 32) atomicAdd(&ar[f], hr[f] * alpha);
}

__global__ void gat_finalize(float* __restrict__ io, const float* __restrict__ bias,
                             long long total, int F, int do_relu) {
  const long long i = (long long)blockIdx.x * blockDim.x + threadIdx.x;
  if (i >= total) return;
  const int f = (int)(i % F);
  float v = io[i] + bias[f];
  io[i] = do_relu ? fmaxf(v, 0.0f) : v;
}

// ---------------------------------------------------------------------------
extern "C" void kernel_launch(void* const* d_in, const int* in_sizes, int n_in,
                              void* d_out, int out_size, void* d_ws, size_t ws_size,
                              hipStream_t stream) {
  const float*     x   = (const float*)d_in[0];
  const long long* ei  = (const long long*)d_in[1];  // int64 [2, E]
  const float*     W1  = (const float*)d_in[2];
  const float*     as1 = (const float*)d_in[3];
  const float*     ad1 = (const float*)d_in[4];
  const float*     b1  = (const float*)d_in[5];
  const float*     W2  = (const float*)d_in[6];
  const float*     as2 = (const float*)d_in[7];
  const float*     ad2 = (const float*)d_in[8];
  const float*     b2  = (const float*)d_in[9];
  float* out = (float*)d_out;

  const int H    = in_sizes[3];            // 64
  const int Cc   = in_sizes[7];            // 40
  const int Ccp  = 48;                     // padded to 3 WMMA N-tiles
  const int Fin  = in_sizes[2] / H;        // 128
  const int N    = in_sizes[0] / Fin;      // 100000
  const long long E = in_sizes[1] / 2;     // 1600000
  const long long EN = E + N;

  // Workspace layout (floats)
  float* ws = (float*)d_ws;
  size_t off = 0;
  float* h1   = ws + off; off += (size_t)N * H;    // x @ W1           [N x 64]
  float* agg1 = ws + off; off += (size_t)N * H;    // layer-1 out      [N x 64]
  float* h2   = ws + off; off += (size_t)N * Ccp;  // agg1 @ W2 padded [N x 48]
  float* a_s  = ws + off; off += (size_t)N;
  float* a_d  = ws + off; off += (size_t)N;
  float* mx   = ws + off; off += (size_t)N;
  float* dn   = ws + off; off += (size_t)N;
  float* p    = ws + off; off += (size_t)EN;
  (void)off; (void)ws_size; (void)n_in; (void)out_size;

  const int TB  = 256;                     // 8 wave32s per block
  const int WPB = TB / 32;
  const int mtiles = (N + 15) / 16;
  dim3 gGemm((unsigned)((mtiles + WPB - 1) / WPB));
  dim3 gNode((unsigned)((N + TB - 1) / TB));
  dim3 gEdge((unsigned)((EN + TB - 1) / TB));
  dim3 gAgg((unsigned)((EN * 32 + TB - 1) / TB));
  dim3 gFin1((unsigned)(((size_t)N * H + TB - 1) / TB));
  dim3 gFin2((unsigned)(((size_t)N * Cc + TB - 1) / TB));

  // ---------------- layer 1 (F_out = 64 -> 4 N-tiles, no padding) --------
  gat_gemm_wmma<4, 128><<<gGemm, TB, 0, stream>>>(x, W1, h1, N, Fin, H);
  gat_node_prep<<<gNode, TB, 0, stream>>>(h1, H, as1, ad1, a_s, a_d, mx, dn,
                                          agg1, N, H);
  gat_edge_max<<<gEdge, TB, 0, stream>>>(ei, a_s, a_d, mx, E, N);
  gat_edge_exp<<<gEdge, TB, 0, stream>>>(ei, a_s, a_d, mx, dn, p, E, N);
  gat_edge_aggregate<<<gAgg, TB, 0, stream>>>(ei, p, dn, h1, H, agg1, E, N, H);
  gat_finalize<<<gFin1, TB, 0, stream>>>(agg1, b1, (long long)N * H, H, 1);

  // ---------------- layer 2 (F_out = 40 -> 3 N-tiles, padded to 48) ------
  gat_gemm_wmma<3, 64><<<gGemm, TB, 0, stream>>>(agg1, W2, h2, N, H, Cc);
  gat_node_prep<<<gNode, TB, 0, stream>>>(h2, Ccp, as2, ad2, a_s, a_d, mx, dn,
                                          out, N, Cc);
  gat_edge_max<<<gEdge, TB, 0, stream>>>(ei, a_s, a_d, mx, E, N);
  gat_edge_exp<<<gEdge, TB, 0, stream>>>(ei, a_s, a_d, mx, dn, p, E, N);
  gat_edge_aggregate<<<gAgg, TB, 0, stream>>>(ei, p, dn, h2, Ccp, out, E, N, Cc);
  gat_finalize<<<gFin2, TB, 0, stream>>>(out, b2, (long long)N * Cc, Cc, 0);
}